// Mamba2Block_25563645346592
// MI455X (gfx1250) — compile-verified
//
#include <hip/hip_runtime.h>
#include <hip/hip_bf16.h>
#include <math.h>

// ---------------------------------------------------------------------------
// Mamba block forward for MI455X (gfx1250, wave32, WMMA).
// Pipeline:
//   1) gemm_f32_wmma<4> : xz = x @ W_in.T                        [2048 x 2048]
//   2) conv_silu        : causal depthwise conv + SiLU -> xs     [2048 x 1024]
//   3) gemm_f32_wmma<1> : dbc = xs @ W_x.T                       [2048 x  160]
//   4) gemm_f32_wmma<4> : delta = softplus(dt @ W_dt.T + b_dt)   [2048 x 1024]
//   5) mamba_scan       : selective scan + skip + SiLU gate -> y [2048 x 1024]
//   6) gemm_f32_wmma<4> : out = y @ W_out.T                      [2048 x  512]
// dA = exp(delta*A) is recomputed inside the scan (never materialized:
// avoids 2 x 536 MB of [B,L,D,N] traffic the reference implies).
// ---------------------------------------------------------------------------

#define B_SZ     2
#define L_SZ     1024
#define DMODEL   512
#define DINNER   1024
#define NSTATE   64
#define DTRANK   32
#define DBC_W    (DTRANK + 2 * NSTATE)   // 160
#define ROWS     (B_SZ * L_SZ)           // 2048

typedef float v2f __attribute__((ext_vector_type(2)));
typedef float v8f __attribute__((ext_vector_type(8)));

__device__ __forceinline__ float silu_f(float v) {
    return v / (1.0f + __expf(-v));
}

// ---------------------------------------------------------------------------
// fp32 WMMA GEMM: Out[M,N] = X[M,K] @ W[N,K]^T  (row-major), X row stride ldx.
// Each wave computes a 16 x (16*NT) strip; the A fragment is loaded once per
// k-step and reused across NT wmma ops (1 + NT loads per NT wmma).
// Ping-pong distance-2 pipeline (unroll-1 pair loop, two static stages):
//   iter(k): wmma(st0=k-8); st0 <- load k; wmma(st1=k-4); st1 <- load k+4
// Each WMMA batch has exactly NT+1 *younger* loads outstanding
// (s_wait_loadcnt <= NT+1) and each stage's loads get a half-iteration of
// latency cover. No rotation copies: stages are constant-indexed.
// epilogue: 0 = none, 1 = softplus(acc + bias[col]).
//
// v_wmma_f32_16x16x4_f32 layouts (ISA 7.12.2):
//   A 16x4 : lanes 0-15 M=lane (K0,K1), lanes 16-31 M=lane-16 (K2,K3)
//   B 4x16 : mirrored over N
//   C/D    : VGPR r -> M = r + 8*(lane>=16), N = lane & 15
// ---------------------------------------------------------------------------
template <int NT>
__global__ void gemm_f32_wmma(const float* __restrict__ X, int ldx,
                              const float* __restrict__ W,   // [N,K], ld = K
                              float* __restrict__ Out,       // [M,N]
                              int M, int N, int K,
                              const float* __restrict__ bias, int epilogue)
{
    const int lane   = threadIdx.x & 31;
    const int wave   = threadIdx.x >> 5;
    const int tile   = blockIdx.x * (blockDim.x >> 5) + wave;
    const int tilesN = N / (16 * NT);
    const int m0 = (tile / tilesN) << 4;
    const int n0 = (tile % tilesN) * (16 * NT);
    if (m0 >= M) return;

    const int half = lane >> 4;        // 0: K0/K1 ; 1: K2/K3
    const int l15  = lane & 15;
    const int koff = half << 1;

    const float* __restrict__ xptr = X + (size_t)(m0 + l15) * ldx + koff;
    const float* wptr[NT];
#pragma unroll
    for (int j = 0; j < NT; ++j)
        wptr[j] = W + (size_t)(n0 + j * 16 + l15) * K + koff;

    v8f c[NT];
#pragma unroll
    for (int j = 0; j < NT; ++j) c[j] = (v8f){};

    // ---- prologue: fill both pipeline stages (k = 0 and k = 4) ----
    v2f a_st[2];
    v2f b_st[2][NT];
    a_st[0] = *reinterpret_cast<const v2f*>(xptr);
#pragma unroll
    for (int j = 0; j < NT; ++j)
        b_st[0][j] = *reinterpret_cast<const v2f*>(wptr[j]);
    a_st[1] = *reinterpret_cast<const v2f*>(xptr + 4);
#pragma unroll
    for (int j = 0; j < NT; ++j)
        b_st[1][j] = *reinterpret_cast<const v2f*>(wptr[j] + 4);

    // ---- ping-pong main loop: consume a stage, then refill it with k+8 ----
#pragma unroll 1
    for (int k = 8; k <= K - 8; k += 8) {
#pragma unroll
        for (int j = 0; j < NT; ++j)
            // (neg_a, A, neg_b, B, c_mod, C, reuse_a, reuse_b)
            c[j] = __builtin_amdgcn_wmma_f32_16x16x4_f32(false, a_st[0], false,
                                                         b_st[0][j], (short)0,
                                                         c[j], false, false);
        a_st[0] = *reinterpret_cast<const v2f*>(xptr + k);
#pragma unroll
        for (int j = 0; j < NT; ++j)
            b_st[0][j] = *reinterpret_cast<const v2f*>(wptr[j] + k);

#pragma unroll
        for (int j = 0; j < NT; ++j)
            c[j] = __builtin_amdgcn_wmma_f32_16x16x4_f32(false, a_st[1], false,
                                                         b_st[1][j], (short)0,
                                                         c[j], false, false);
        a_st[1] = *reinterpret_cast<const v2f*>(xptr + k + 4);
#pragma unroll
        for (int j = 0; j < NT; ++j)
            b_st[1][j] = *reinterpret_cast<const v2f*>(wptr[j] + k + 4);
    }

    // ---- drain both stages ----
#pragma unroll
    for (int j = 0; j < NT; ++j)
        c[j] = __builtin_amdgcn_wmma_f32_16x16x4_f32(false, a_st[0], false,
                                                     b_st[0][j], (short)0,
                                                     c[j], false, false);
#pragma unroll
    for (int j = 0; j < NT; ++j)
        c[j] = __builtin_amdgcn_wmma_f32_16x16x4_f32(false, a_st[1], false,
                                                     b_st[1][j], (short)0,
                                                     c[j], false, false);

    // ---- epilogue + store ----
#pragma unroll
    for (int j = 0; j < NT; ++j) {
#pragma unroll
        for (int r = 0; r < 8; ++r) {
            const int row = m0 + r + (half << 3);
            const int col = n0 + j * 16 + l15;
            float v = c[j][r];
            if (epilogue == 1) {
                v += bias[col];
                v = (v > 20.0f) ? v : log1pf(__expf(v));   // softplus
            }
            Out[(size_t)row * N + col] = v;
        }
    }
}

// ---------------------------------------------------------------------------
// Causal depthwise conv1d (taps=4) + bias + SiLU.
// xz is [ROWS, 2*DINNER]; the SSM half is columns [0, DINNER).
// ---------------------------------------------------------------------------
__global__ void conv_silu(const float* __restrict__ xz,
                          const float* __restrict__ conv_w,   // [DINNER,1,4]
                          const float* __restrict__ conv_b,   // [DINNER]
                          float* __restrict__ xs)
{
    const int idx = blockIdx.x * blockDim.x + threadIdx.x;
    if (idx >= ROWS * DINNER) return;
    const int d = idx & (DINNER - 1);
    const int t = idx >> 10;            // b*L + l
    const int l = t & (L_SZ - 1);

    float acc = conv_b[d];
#pragma unroll
    for (int k = 0; k < 4; ++k) {
        const int ls = l + k - 3;       // pad (3,0): out[l] = sum_k x[l+k-3]*w[k]
        if (ls >= 0)
            acc = fmaf(conv_w[d * 4 + k],
                       xz[(size_t)(t + k - 3) * (2 * DINNER) + d], acc);
    }
    xs[idx] = silu_f(acc);
}

// ---------------------------------------------------------------------------
// Selective scan. One wave32 per (b, d): lane owns states n=lane and n=lane+32.
// Per step: h = exp(delta*A)*h + delta*x*B ; y = wave_sum(h*C) + D*x ; *silu(z)
// ---------------------------------------------------------------------------
__global__ void mamba_scan(const float* __restrict__ delta,  // [ROWS, DINNER]
                           const float* __restrict__ xs,     // [ROWS, DINNER]
                           const float* __restrict__ dbc,    // [ROWS, 160]
                           const float* __restrict__ xz,     // [ROWS, 2048] (z half)
                           const float* __restrict__ A_log,  // [64]
                           const float* __restrict__ D_param,// [DINNER]
                           float* __restrict__ y)            // [ROWS, DINNER]
{
    const int lane = threadIdx.x & 31;
    const int wave = (blockIdx.x * blockDim.x + threadIdx.x) >> 5;
    if (wave >= B_SZ * DINNER) return;
    const int b = wave >> 10;           // / DINNER
    const int d = wave & (DINNER - 1);

    const float A0 = -__expf(A_log[lane]);
    const float A1 = -__expf(A_log[lane + 32]);
    const float Dd = D_param[d];

    float h0 = 0.0f, h1 = 0.0f;
    for (int l = 0; l < L_SZ; ++l) {
        const size_t t    = (size_t)b * L_SZ + l;
        const float  dv   = delta[t * DINNER + d];   // wave-uniform
        const float  xv   = xs[t * DINNER + d];      // wave-uniform
        const float* row  = dbc + t * DBC_W;
        const float  Bn0  = row[DTRANK + lane];
        const float  Bn1  = row[DTRANK + 32 + lane];
        const float  Cn0  = row[DTRANK + NSTATE + lane];
        const float  Cn1  = row[DTRANK + NSTATE + 32 + lane];

        // Warm L0 for the shared B/C row a few steps ahead (global_prefetch_b8)
        if (l + 8 < L_SZ)
            __builtin_prefetch(dbc + (t + 8) * DBC_W + DTRANK + lane, 0, 0);

        const float dbx = dv * xv;
        h0 = __expf(dv * A0) * h0 + dbx * Bn0;
        h1 = __expf(dv * A1) * h1 + dbx * Bn1;

        float p = h0 * Cn0 + h1 * Cn1;
#pragma unroll
        for (int off = 16; off > 0; off >>= 1)
            p += __shfl_xor(p, off, 32);             // butterfly: all lanes get sum

        if (lane == 0) {
            const float zv = xz[t * (2 * DINNER) + DINNER + d];
            y[t * DINNER + d] = (p + Dd * xv) * silu_f(zv);
        }
    }
}

// ---------------------------------------------------------------------------
// Host-side launch. Inputs (setup_inputs order):
// 0:x 1:W_in 2:conv_w 3:conv_b 4:W_x 5:W_dt 6:b_dt 7:A_log 8:D_param 9:W_out
// ---------------------------------------------------------------------------
extern "C" void kernel_launch(void* const* d_in, const int* in_sizes, int n_in,
                              void* d_out, int out_size, void* d_ws, size_t ws_size,
                              hipStream_t stream)
{
    const float* x      = (const float*)d_in[0];
    const float* W_in   = (const float*)d_in[1];
    const float* conv_w = (const float*)d_in[2];
    const float* conv_b = (const float*)d_in[3];
    const float* W_x    = (const float*)d_in[4];
    const float* W_dt   = (const float*)d_in[5];
    const float* b_dt   = (const float*)d_in[6];
    const float* A_log  = (const float*)d_in[7];
    const float* D_par  = (const float*)d_in[8];
    const float* W_out  = (const float*)d_in[9];
    float* out = (float*)d_out;

    // Workspace layout (floats), ~41 MB total
    float* ws    = (float*)d_ws;
    float* xz    = ws;                                   // ROWS * 2048
    float* xs    = xz    + (size_t)ROWS * 2 * DINNER;    // ROWS * 1024
    float* dbc   = xs    + (size_t)ROWS * DINNER;        // ROWS * 160
    float* delta = dbc   + (size_t)ROWS * DBC_W;         // ROWS * 1024
    float* ybuf  = delta + (size_t)ROWS * DINNER;        // ROWS * 1024

    const int BLK = 256;                 // 8 waves / block
    const int WPB = BLK / 32;

    // 1) xz = x @ W_in^T           M=2048 N=2048 K=512   (NT=4 strips)
    {
        int tiles = (ROWS / 16) * ((2 * DINNER) / 64);   // 128*32 = 4096
        gemm_f32_wmma<4><<<tiles / WPB, BLK, 0, stream>>>(
            x, DMODEL, W_in, xz, ROWS, 2 * DINNER, DMODEL, nullptr, 0);
    }
    // 2) depthwise conv + SiLU -> xs
    conv_silu<<<(ROWS * DINNER) / BLK, BLK, 0, stream>>>(xz, conv_w, conv_b, xs);

    // 3) dbc = xs @ W_x^T          M=2048 N=160 K=1024   (NT=1: N not /64)
    {
        int tiles = (ROWS / 16) * (DBC_W / 16);          // 128*10 = 1280
        gemm_f32_wmma<1><<<tiles / WPB, BLK, 0, stream>>>(
            xs, DINNER, W_x, dbc, ROWS, DBC_W, DINNER, nullptr, 0);
    }
    // 4) delta = softplus(dt @ W_dt^T + b_dt)   M=2048 N=1024 K=32
    //    dt is columns [0,32) of dbc -> X = dbc with ldx = 160.
    {
        int tiles = (ROWS / 16) * (DINNER / 64);         // 128*16 = 2048
        gemm_f32_wmma<4><<<tiles / WPB, BLK, 0, stream>>>(
            dbc, DBC_W, W_dt, delta, ROWS, DINNER, DTRANK, b_dt, 1);
    }
    // 5) selective scan (+skip +gate) -> ybuf   [2048 waves]
    {
        int waves = B_SZ * DINNER;
        mamba_scan<<<(waves * 32) / BLK, BLK, 0, stream>>>(
            delta, xs, dbc, xz, A_log, D_par, ybuf);
    }
    // 6) out = y @ W_out^T         M=2048 N=512 K=1024
    {
        int tiles = (ROWS / 16) * (DMODEL / 64);         // 128*8 = 1024
        gemm_f32_wmma<4><<<tiles / WPB, BLK, 0, stream>>>(
            ybuf, DINNER, W_out, out, ROWS, DMODEL, DINNER, nullptr, 0);
    }
}